// ShapeBlock_34299608826088
// MI455X (gfx1250) — compile-verified
//
#include <hip/hip_runtime.h>
#include <math.h>

// Problem constants (from reference)
#define B_ROWS   4096
#define C_CH     8
#define LEN_TS   1024
#define DIM_SEL  2
#define T_LEN    628          // END - START
#define L_LEN    64
#define NW       565          // T_LEN - L_LEN + 1
#define E_DIM    128
#define INV_NORM (1.0f / 1000.0f)
#define MAX_CI   3.0f

typedef float v2f __attribute__((ext_vector_type(2)));
typedef float v8f __attribute__((ext_vector_type(8)));

// ---------------------------------------------------------------------------
// Kernel 1: per batch row — fused CID scan + argmin + best-window gather.
// One block (256 thr = 8 waves) per row; row cached in LDS (2.5 KB).
// ---------------------------------------------------------------------------
__global__ __launch_bounds__(256)
void cid_argmin_kernel(const float* __restrict__ x,
                       const float* __restrict__ shapelet,
                       float* __restrict__ bestwin) {
    __shared__ float sm_piss[T_LEN];
    __shared__ float sm_s[L_LEN];
    __shared__ float red_val[256];
    __shared__ int   red_idx[256];

    const int b   = blockIdx.x;
    const int tid = threadIdx.x;
    const float* xrow = x + (size_t)b * (C_CH * LEN_TS) + DIM_SEL * LEN_TS; // START=0

    for (int i = tid; i < T_LEN; i += 256) sm_piss[i] = xrow[i];
    if (tid < L_LEN) sm_s[tid] = shapelet[tid];
    __syncthreads();

    // shapelet complexity (redundant per-thread; 63 FMAs, negligible)
    float ssum = 0.f;
#pragma unroll
    for (int k = 0; k < L_LEN - 1; ++k) {
        float d = sm_s[k + 1] - sm_s[k];
        ssum = fmaf(d, d, ssum);
    }
    const float s_ci = sqrtf(ssum + INV_NORM);

    float best = 3.4e38f;
    int   bidx = 0x7fffffff;
    for (int j = tid; j < NW; j += 256) {
        float prev = sm_piss[j];
        float d0   = prev - sm_s[0];
        float ed2  = d0 * d0;
        float dd2  = 0.f;
#pragma unroll
        for (int k = 1; k < L_LEN; ++k) {
            float v  = sm_piss[j + k];
            float d  = v - sm_s[k];
            ed2 = fmaf(d, d, ed2);
            float dd = v - prev;
            dd2 = fmaf(dd, dd, dd2);
            prev = v;
        }
        float pci = sqrtf(dd2 + INV_NORM);
        float cf  = fminf(fmaxf(pci, s_ci) / fminf(pci, s_ci), MAX_CI);
        float cid = sqrtf(ed2) * cf;
        // within a thread j strictly increases, so '<' keeps first occurrence
        if (cid < best) { best = cid; bidx = j; }
    }

    red_val[tid] = best;
    red_idx[tid] = bidx;
    __syncthreads();
    // block argmin; tie -> smaller index (matches jnp.argmin first-occurrence)
    for (int s = 128; s > 0; s >>= 1) {
        if (tid < s) {
            float v2 = red_val[tid + s];
            int   i2 = red_idx[tid + s];
            if (v2 < red_val[tid] || (v2 == red_val[tid] && i2 < red_idx[tid])) {
                red_val[tid] = v2;
                red_idx[tid] = i2;
            }
        }
        __syncthreads();
    }
    const int widx = red_idx[0];
    if (tid < L_LEN) bestwin[(size_t)b * L_LEN + tid] = sm_piss[widx + tid];
}

// ---------------------------------------------------------------------------
// Kernel 2: combined bias  c[e] = b1[e] - (shapelet @ W2.T + b2)[e]
// ---------------------------------------------------------------------------
__global__ __launch_bounds__(128)
void bias_kernel(const float* __restrict__ shapelet,
                 const float* __restrict__ W2,
                 const float* __restrict__ b1,
                 const float* __restrict__ b2,
                 float* __restrict__ cbias) {
    const int e = threadIdx.x;   // 0..127
    float acc = 0.f;
#pragma unroll
    for (int k = 0; k < L_LEN; ++k)
        acc = fmaf(shapelet[k], W2[e * L_LEN + k], acc);
    cbias[e] = b1[e] - (acc + b2[e]);
}

// ---------------------------------------------------------------------------
// Kernel 3: out[b,e] = sum_k bestwin[b,k] * W1[e,k] + cbias[e]
// M=4096, N=128, K=64 fp32 GEMM via V_WMMA_F32_16X16X4_F32 (exact f32 FMA).
// One wave per 16x16 tile; 8 waves/block cover all N; 256 blocks cover M.
// Fragment layouts per CDNA5 ISA 7.12.2 (32-bit A 16x4, B 4x16, C/D 16x16).
// ---------------------------------------------------------------------------
__global__ __launch_bounds__(256)
void wmma_gemm_kernel(const float* __restrict__ A,     // 4096 x 64, row major
                      const float* __restrict__ W1,    // 128 x 64, row major (B[k,e]=W1[e,k])
                      const float* __restrict__ cbias, // 128
                      float* __restrict__ out) {       // 4096 x 128
    const int lane = threadIdx.x & 31;
    const int wave = threadIdx.x >> 5;      // 0..7 -> N tile
    const int m0   = blockIdx.x * 16;
    const int n0   = wave * 16;
    const int half = lane >> 4;             // 0: lanes 0-15, 1: lanes 16-31
    const int l15  = lane & 15;

    // A frag: lane holds row (m0+l15); v0 = K(4kk + 2*half), v1 = +1
    const v2f* ap = reinterpret_cast<const v2f*>(A  + (size_t)(m0 + l15) * L_LEN + 2 * half);
    // B frag: lane holds col (n0+l15); v0 = K(4kk + 2*half), v1 = +1
    const v2f* bp = reinterpret_cast<const v2f*>(W1 + (size_t)(n0 + l15) * L_LEN + 2 * half);

    v8f acc = {};
#pragma unroll
    for (int kk = 0; kk < L_LEN / 4; ++kk) {   // 16 WMMA ops, K=4 each
        v2f a = ap[2 * kk];
        v2f b = bp[2 * kk];
        acc = __builtin_amdgcn_wmma_f32_16x16x4_f32(
            /*neg_a=*/false, a, /*neg_b=*/false, b,
            /*c_mod=*/(short)0, acc, /*reuse_a=*/false, /*reuse_b=*/false);
    }

    const float cb = cbias[n0 + l15];
#pragma unroll
    for (int r = 0; r < 8; ++r) {
        const int m = m0 + r + half * 8;      // C/D: VGPR r -> row r (lanes 0-15) / r+8
        out[(size_t)m * E_DIM + n0 + l15] = acc[r] + cb;
    }
}

// ---------------------------------------------------------------------------
extern "C" void kernel_launch(void* const* d_in, const int* in_sizes, int n_in,
                              void* d_out, int out_size, void* d_ws, size_t ws_size,
                              hipStream_t stream) {
    const float* x        = (const float*)d_in[0];
    const float* shapelet = (const float*)d_in[1];
    const float* W1       = (const float*)d_in[2];
    const float* b1       = (const float*)d_in[3];
    const float* W2       = (const float*)d_in[4];
    const float* b2       = (const float*)d_in[5];
    float* out = (float*)d_out;

    float* bestwin = (float*)d_ws;                       // 4096*64 floats = 1 MB
    float* cbias   = bestwin + (size_t)B_ROWS * L_LEN;   // 128 floats

    cid_argmin_kernel<<<B_ROWS, 256, 0, stream>>>(x, shapelet, bestwin);
    bias_kernel<<<1, 128, 0, stream>>>(shapelet, W2, b1, b2, cbias);
    wmma_gemm_kernel<<<B_ROWS / 16, 256, 0, stream>>>(bestwin, W1, cbias, out);
}